// QAEModel_22419729285342
// MI455X (gfx1250) — compile-verified
//
#include <hip/hip_runtime.h>

// CDNA5 / gfx1250 — wave32, WMMA 16x16x4 FP32, 320KB LDS per WGP, TDM staging.

typedef __attribute__((ext_vector_type(2))) float    v2f;
typedef __attribute__((ext_vector_type(8))) float    v8f;
typedef __attribute__((ext_vector_type(4))) unsigned v4u;
typedef __attribute__((ext_vector_type(8))) unsigned v8u;

#define NQ    8
#define DIM   256          // 2^NQ
#define MLAY  40
#define BATCH 16384
#define USE_TDM 1

// U is stored (in global scratch AND in LDS) in a K-pair interleaved layout:
//   element (k, n)  ->  (k>>1)*512 + 2*n + (k&1)
// so a WMMA B-fragment (rows k, k+1 at col n) is one aligned 8-byte load.

// ---------------------------------------------------------------------------
// Kernel 1: build the 256x256 circuit matrix U from theta.
// One wave per basis row; the row's 256-amplitude state lives in LDS.
// Wave32 lockstep + in-order LDS => no barriers needed inside a wave.
// ---------------------------------------------------------------------------
__global__ void __launch_bounds__(256)
build_u(const float* __restrict__ theta, float* __restrict__ Uout) {
    extern __shared__ float sm[];
    float* cs   = sm;          // 328 * 2 cos/sin values (shared by all waves)
    float* rows = sm + 672;    // 8 waves * 256 floats of row state

    const int tid  = threadIdx.x;
    const int wave = tid >> 5;
    const int lane = tid & 31;

    // Precompute cos/sin(theta/2) once per block.
    for (int g = tid; g < (MLAY + 1) * NQ; g += 256) {
        float th = 0.5f * theta[g];
        cs[2 * g]     = cosf(th);
        cs[2 * g + 1] = sinf(th);
    }
    __syncthreads();

    const int r = blockIdx.x * 8 + wave;   // basis row this wave evolves
    float* st = rows + (wave << 8);

    // state = e_r
    #pragma unroll
    for (int v = 0; v < 8; ++v) {
        int i = (v << 5) | lane;
        st[i] = (i == r) ? 1.0f : 0.0f;
    }

    for (int m = 0; m <= MLAY; ++m) {
        // ---- RY layer on all 8 qubits ----
        for (int q = 0; q < NQ; ++q) {
            const int g = m * NQ + q;
            const float c = cs[2 * g];
            const float s = cs[2 * g + 1];
            const int pos    = 7 - q;       // bit position of qubit q
            const int stride = 1 << pos;
            const int lowm   = stride - 1;
            #pragma unroll
            for (int it = 0; it < 4; ++it) {          // 128 pairs / 32 lanes
                int p  = (it << 5) | lane;
                int i0 = ((p >> pos) << (pos + 1)) | (p & lowm);
                int i1 = i0 | stride;
                float a0 = st[i0], a1 = st[i1];
                // out_j = sum_i psi_i * G[i][j], G = [[c,-s],[s,c]]
                st[i0] = c * a0 + s * a1;
                st[i1] = c * a1 - s * a0;
            }
        }
        // ---- ring CNOT layer (forward iff m even), skipped after last RY ----
        if (m < MLAY) {
            const bool fwd = ((m & 1) == 0);
            for (int q = 0; q < NQ; ++q) {
                const int tgt  = fwd ? ((q + 1) & 7) : ((q + 7) & 7);
                const int cpos = 7 - q, tpos = 7 - tgt;
                const int cbit = 1 << cpos, tbit = 1 << tpos;
                const int p1 = (cpos < tpos) ? cpos : tpos;
                const int p2 = (cpos < tpos) ? tpos : cpos;
                #pragma unroll
                for (int it = 0; it < 2; ++it) {      // 64 swap pairs / 32 lanes
                    int p    = (it << 5) | lane;
                    int low  = p & ((1 << p1) - 1);
                    int mid  = (p >> p1) & ((1 << (p2 - p1 - 1)) - 1);
                    int high = p >> (p2 - 1);
                    int col  = low | (mid << (p1 + 1)) | (high << (p2 + 1));
                    col |= cbit;                      // control = 1, target = 0
                    float a = st[col], b = st[col | tbit];
                    st[col] = b;
                    st[col | tbit] = a;
                }
            }
        }
    }

    // Write this row of U in the K-pair interleaved layout (row k = r).
    const int base = ((r >> 1) << 9) + (r & 1);
    #pragma unroll
    for (int v = 0; v < 8; ++v) {
        int i = (v << 5) | lane;
        Uout[base + (i << 1)] = st[i];
    }
}

// ---------------------------------------------------------------------------
// Kernel 2: Y = X @ U via v_wmma_f32_16x16x4_f32, fused sum-of-squares
// epilogue. Whole U (256 KB) staged in LDS via the Tensor Data Mover
// (one tensor_load_to_lds per wave, 32 KB slab each).
// ---------------------------------------------------------------------------
__global__ void __launch_bounds__(256)
gemm_sq(const float* __restrict__ X, const float* __restrict__ U,
        float* __restrict__ out) {
    extern __shared__ float Ulds[];     // 65536 floats = 256 KB

    const int tid  = threadIdx.x;
    const int wave = tid >> 5;
    const int lane = tid & 31;

#if USE_TDM
    {
        // Each wave DMAs a contiguous 32 KB slab (described as a 2D tile of
        // 256 x 32 4-byte elements, stride 256) from global U into LDS.
        unsigned lds_off  = (unsigned)(size_t)(void*)Ulds + (unsigned)(wave << 15);
        unsigned long long ga = (unsigned long long)(size_t)(U + (wave << 13));
        // D# group 0: count=1 | lds_addr | global_addr[56:0] | type=2 (image)
        v4u g0;
        g0.x = 1u;                              // count=1, is_restore=0
        g0.y = lds_off;                         // lds_addr (bytes)
        g0.z = (unsigned)(ga & 0xffffffffu);    // global_addr[31:0]
        g0.w = (unsigned)(ga >> 32) | (2u << 30); // global_addr[56:32] | type=2
        // D# group 1
        v8u g1;
        g1[0] = 0x00020000u;          // workgroup_mask=0, data_size=2 (4B)
        g1[1] = (unsigned)(DIM) << 16; // tensor_dim0[15:0] @ bits 63:48
        g1[2] = 32u << 16;             // tensor_dim1[15:0]=32 @ bits 95:80
        g1[3] = (unsigned)(DIM) << 16; // tile_dim0=256 @ bits 127:112
        g1[4] = 32u;                   // tile_dim1=32 @ bits 143:128
        g1[5] = (unsigned)(DIM);       // tensor_dim0_stride[31:0]=256
        g1[6] = 0u;
        g1[7] = 0u;
        asm volatile("tensor_load_to_lds %0, %1"
                     :: "s"(g0), "s"(g1) : "memory");
        __builtin_amdgcn_s_wait_tensorcnt(0);
    }
    __syncthreads();   // other waves' slabs must be visible too
#else
    {
        const float4* src = (const float4*)U;
        float4*       dst = (float4*)Ulds;
        for (int i = tid; i < (DIM * DIM) / 4; i += 256) dst[i] = src[i];
    }
    __syncthreads();
#endif

    const int rowBase = (blockIdx.x * 8 + wave) * 16;
    const int ln = lane & 15;        // sub-row / sub-col within tile
    const int kh = lane >> 4;        // K half: lanes 0-15 -> K0,K1; 16-31 -> K2,K3
    const int kk = kh << 1;

    v8f acc[16];
    #pragma unroll
    for (int nt = 0; nt < 16; ++nt) acc[nt] = (v8f)0.0f;

    const float* xrow = X + (size_t)(rowBase + ln) * DIM + kk;

    #pragma unroll 4
    for (int ks = 0; ks < DIM / 4; ++ks) {
        // A fragment (16x4 f32): lane ln holds row M=ln, K = 4ks+kk, 4ks+kk+1
        v2f a = *(const v2f*)(xrow + (ks << 2));
        // B fragments (4x16 f32) from interleaved LDS: one b64 per fragment.
        // Row pair (4ks+kk, 4ks+kk+1) lives at block (2ks + kh)*512.
        const v2f* bp = (const v2f*)(Ulds + (((ks << 1) + kh) << 9)) + ln;
        #pragma unroll
        for (int nt = 0; nt < 16; ++nt) {
            v2f b = bp[nt << 4];     // {U[k][n], U[k+1][n]} in one load
            acc[nt] = __builtin_amdgcn_wmma_f32_16x16x4_f32(
                false, a, false, b, (short)0, acc[nt], false, false);
        }
    }

    // Epilogue: out[row, l] = sum over cols ≡ l (mod 8) of Y².
    // D layout: VGPR v, lanes 0-15: (M=v, N=lane); lanes 16-31: (M=v+8, N=lane-16).
    float ps[8];
    #pragma unroll
    for (int v = 0; v < 8; ++v) {
        float s = 0.0f;
        #pragma unroll
        for (int nt = 0; nt < 16; ++nt) {
            float y = acc[nt][v];
            s += y * y;              // col%8 == lane&7 for every nt
        }
        ps[v] = s;
    }
    // Combine lane pairs (n and n+8 share the same col%8).
    #pragma unroll
    for (int v = 0; v < 8; ++v) ps[v] += __shfl_xor(ps[v], 8, 32);

    if ((lane & 8) == 0) {
        const int rofs = (lane >> 4) << 3;   // 0 for M=0..7, 8 for M=8..15
        const int cm   = lane & 7;
        #pragma unroll
        for (int v = 0; v < 8; ++v)
            out[(size_t)(rowBase + rofs + v) * 8 + cm] = ps[v];
    }
}

// ---------------------------------------------------------------------------
extern "C" void kernel_launch(void* const* d_in, const int* in_sizes, int n_in,
                              void* d_out, int out_size, void* d_ws, size_t ws_size,
                              hipStream_t stream) {
    const float* X     = (const float*)d_in[0];   // [16384, 256]
    const float* theta = (const float*)d_in[1];   // [328]
    float*       out   = (float*)d_out;           // [16384, 8]
    float*       Uw    = (float*)d_ws;            // 256*256 floats scratch

    // Kernel 1: 256 rows, one wave each -> 32 blocks * 8 waves.
    size_t sm1 = (672 + 8 * DIM) * sizeof(float);
    build_u<<<32, 256, sm1, stream>>>(theta, Uw);

    // Kernel 2: 16384 rows / (8 waves * 16 rows) = 128 blocks; U in 256KB LDS.
    size_t sm2 = DIM * DIM * sizeof(float);
    gemm_sq<<<BATCH / 128, 256, sm2, stream>>>(X, Uw, out);
}